// MTGCN_62921270886524
// MI455X (gfx1250) — compile-verified
//
#include <hip/hip_runtime.h>

#define N_NODES 50000
#define IN_DIM  128
#define H1      300
#define H2      100
#define OUT_DIM 2
#define NE      800000
#define NP      400000

typedef __attribute__((ext_vector_type(2))) float v2f;
typedef __attribute__((ext_vector_type(4))) float v4f;
typedef __attribute__((ext_vector_type(8))) float v8f;

__device__ __forceinline__ v8f wmma_f32(v2f a, v2f b, v8f c) {
    // V_WMMA_F32_16X16X4_F32 : D = A(16x4,f32) * B(4x16,f32) + C(16x16,f32)
    return __builtin_amdgcn_wmma_f32_16x16x4_f32(
        /*neg_a=*/false, a, /*neg_b=*/false, b,
        /*c_mod=*/(short)0, c, /*reuse_a=*/false, /*reuse_b=*/false);
}

// ---------------- utility kernels ----------------

__global__ void zero4_kernel(float* __restrict__ p, int n4) {  // n4 = count of float4
    int i = blockIdx.x * blockDim.x + threadIdx.x;
    if (i < n4) {
        v4f zv = {0.f, 0.f, 0.f, 0.f};
        *(v4f*)(p + 4 * (long long)i) = zv;
    }
}

__global__ void degree_kernel(const int* __restrict__ src, float* __restrict__ deg, int E) {
    int i = blockIdx.x * blockDim.x + threadIdx.x;
    if (i < E) atomicAdd(&deg[src[i]], 1.0f);
}

__global__ void dinv_kernel(float* __restrict__ d, int n) {
    int i = blockIdx.x * blockDim.x + threadIdx.x;
    if (i < n) {
        float v = d[i];
        d[i] = (v > 0.f) ? rsqrtf(v) : 0.f;
    }
}

// tx1[dst] += x[src] * (-dinv[src]*dinv[dst]); one thread = one edge x 4 features
template <int F>
__global__ void scatter_kernel(const int* __restrict__ src, const int* __restrict__ dst,
                               const float* __restrict__ dinv, const float* __restrict__ x,
                               float* __restrict__ tx) {
    constexpr int F4 = F >> 2;
    int t = blockIdx.x * blockDim.x + threadIdx.x;
    if (t >= NE * F4) return;
    int e = t / F4;                 // constant divisor -> mul+shift
    int c = (t - e * F4) << 2;
    int s = src[e], d = dst[e];
    float nrm = -dinv[s] * dinv[d];
    v4f v = *(const v4f*)(x + (long long)s * F + c);
    float* o = tx + (long long)d * F + c;
    atomicAdd(o + 0, v.x * nrm);
    atomicAdd(o + 1, v.y * nrm);
    atomicAdd(o + 2, v.z * nrm);
    atomicAdd(o + 3, v.w * nrm);
}

// ---------------- fused WMMA GEMM ----------------
// out[M,N] = maybe_relu( A0[M,K]@B0 + (DUAL ? A1[M,K]@B1 : 0) + bias[N] ) (+ addend[M,N])
// TRANSB=0: B is [K,N] row-major.  TRANSB=1: B is [N,K] row-major (W^T matmul).
// Block = 5 waves, each wave owns one 16x16 output tile; the shared 16-wide B
// column panel is staged (bounds-checked + zero-padded) into LDS, pre-swizzled
// so each lane reads its (B[k],B[k+1]) fragment with one conflict-free ds_load_b64.
template <bool DUAL, bool TRANSB, bool RELU, bool HAS_ADD, int K>
__global__ void __launch_bounds__(160) wmma_gemm_kernel(
        const float* __restrict__ A0, const float* __restrict__ B0,
        const float* __restrict__ A1, const float* __restrict__ B1,
        const float* __restrict__ bias, const float* __restrict__ addend,
        float* __restrict__ out, int M, int N, int ldOut) {
    __shared__ float sB[(DUAL ? 2 : 1) * K * 16];

    const int nt = blockIdx.y;
    const int col0 = nt << 4;

    // cooperative B-panel stage: sB[(k>>1)*32 + n*2 + (k&1)] = B[k][col0+n]
    for (int idx = threadIdx.x; idx < K * 16; idx += blockDim.x) {
        int k = idx >> 4;
        int n = idx & 15;
        int col = col0 + n;
        int so = ((k >> 1) << 5) + (n << 1) + (k & 1);
        bool cv = col < N;
        sB[so] = cv ? (TRANSB ? B0[(long long)col * K + k] : B0[(long long)k * N + col]) : 0.f;
        if (DUAL)
            sB[K * 16 + so] = cv ? B1[(long long)k * N + col] : 0.f;
    }
    __syncthreads();

    const int lane = threadIdx.x & 31;
    const int wave = threadIdx.x >> 5;
    const int mt = blockIdx.x * 5 + wave;
    if ((mt << 4) >= M) return;               // wave-uniform (after barrier)

    const int row0 = mt << 4;
    const int lid = lane & 15;
    const bool hi = lane >= 16;
    const int koff = hi ? 2 : 0;              // lanes 16-31 hold K=2,3 of the frag
    const int am = row0 + lid;
    const int bn = col0 + lid;
    const bool bnv = bn < N;

    const float* pA0 = A0 + (long long)am * K + koff;
    const float* pA1 = DUAL ? (A1 + (long long)am * K + koff) : nullptr;
    const float* sb0 = sB + (lid << 1) + (hi ? 32 : 0);
    const float* sb1 = sb0 + K * 16;

    v8f acc = {0.f, 0.f, 0.f, 0.f, 0.f, 0.f, 0.f, 0.f};

#pragma unroll 4
    for (int k = 0; k < K; k += 4) {
        v2f a = *(const v2f*)(pA0 + k);
        v2f b = *(const v2f*)(sb0 + (k << 3));     // k*8 floats; the hi-lane +32 (koff*16) folded into sb0
        acc = wmma_f32(a, b, acc);
        if (DUAL) {
            v2f a1 = *(const v2f*)(pA1 + k);
            v2f b1 = *(const v2f*)(sb1 + (k << 3));
            acc = wmma_f32(a1, b1, acc);
        }
    }

    float bv = bnv ? bias[bn] : 0.f;
#pragma unroll
    for (int r = 0; r < 8; ++r) {
        int row = row0 + r + (hi ? 8 : 0);        // C/D layout: lanes 16-31 hold M = r+8
        if (bnv && row < M) {
            float v = acc[r] + bv;
            if (RELU) v = fmaxf(v, 0.f);
            if (HAS_ADD) v += addend[(long long)row * ldOut + bn];
            out[(long long)row * ldOut + bn] = v;
        }
    }
}

// ---------------- link-prediction loss ----------------
__global__ void loss_kernel(const int* __restrict__ ia, const int* __restrict__ ib,
                            const float* __restrict__ z, float* __restrict__ acc,
                            int P, int which) {
    int p = blockIdx.x * blockDim.x + threadIdx.x;
    float term = 0.f;
    if (p < P) {
        const float* za = z + (long long)ia[p] * H2;
        const float* zb = z + (long long)ib[p] * H2;
        float s = 0.f;
#pragma unroll
        for (int c = 0; c < H2; c += 4) {
            v4f a = *(const v4f*)(za + c);
            v4f b = *(const v4f*)(zb + c);
            s += a.x * b.x + a.y * b.y + a.z * b.z + a.w * b.w;
        }
        float sig = 1.f / (1.f + expf(-s));
        term = (which == 0) ? logf(sig + 1e-15f) : logf(1.f - sig + 1e-15f);
    }
    __shared__ float red[256];
    red[threadIdx.x] = term;
    __syncthreads();
    for (int off = 128; off > 0; off >>= 1) {
        if ((int)threadIdx.x < off) red[threadIdx.x] += red[threadIdx.x + off];
        __syncthreads();
    }
    if (threadIdx.x == 0) atomicAdd(&acc[which], red[0]);
}

__global__ void finalize_kernel(const float* __restrict__ acc, const float* __restrict__ c1,
                                const float* __restrict__ c2, float* __restrict__ out_tail) {
    out_tail[0] = -(acc[0] + acc[1]) * (1.0f / (float)NP);  // pos_loss + neg_loss
    out_tail[1] = c1[0];
    out_tail[2] = c2[0];
}

// ---------------- launch ----------------
extern "C" void kernel_launch(void* const* d_in, const int* in_sizes, int n_in,
                              void* d_out, int out_size, void* d_ws, size_t ws_size,
                              hipStream_t stream) {
    const float* x    = (const float*)d_in[0];
    const int*   ei   = (const int*)d_in[1];    // [2, NE]
    const int*   epos = (const int*)d_in[2];    // [2, NP]
    const int*   eneg = (const int*)d_in[3];    // [2, NP]
    const float* W1   = (const float*)d_in[4];  // [2,128,300]
    const float* b1   = (const float*)d_in[5];
    const float* W2   = (const float*)d_in[6];  // [2,300,100]
    const float* b2   = (const float*)d_in[7];
    const float* W3   = (const float*)d_in[8];  // [2,100,2]
    const float* b3   = (const float*)d_in[9];
    const float* l1W  = (const float*)d_in[10]; // [100,128]
    const float* l1b  = (const float*)d_in[11];
    const float* l2W  = (const float*)d_in[12]; // [100,128]
    const float* l2b  = (const float*)d_in[13];
    const float* c1   = (const float*)d_in[14];
    const float* c2   = (const float*)d_in[15];
    float* out = (float*)d_out;

    float* ws   = (float*)d_ws;
    float* dinv = ws;                          // 50000 (reused: deg -> dinv)
    float* acc  = ws + 50016;                  // 2 loss accumulators
    float* tx   = ws + 50032;                  // up to 50000*300
    float* h    = tx + 15000000;               // 50000*300
    float* x1   = h + 15000000;                // 50000*100
    float* x2   = h;                           // reuse (h dead after layer 2)
    float* z    = h + 5000000;

    const int* src = ei;
    const int* dst = ei + NE;

    const int BT = 256;
    const dim3 gblk(160);                      // 5 waves; 3125 M-tiles = 625 blocks exactly
    const int gx = 625;

    // degree -> dinv
    zero4_kernel<<<(50032 / 4 + BT - 1) / BT, BT, 0, stream>>>(ws, 50032 / 4);
    degree_kernel<<<(NE + BT - 1) / BT, BT, 0, stream>>>(src, dinv, NE);
    dinv_kernel<<<(N_NODES + BT - 1) / BT, BT, 0, stream>>>(dinv, N_NODES);

    // ---- layer 1: h = relu(x@W1[0] + tx@W1[1] + b1) ----
    zero4_kernel<<<(N_NODES * IN_DIM / 4 + BT - 1) / BT, BT, 0, stream>>>(tx, N_NODES * IN_DIM / 4);
    scatter_kernel<IN_DIM><<<(NE * (IN_DIM / 4) + BT - 1) / BT, BT, 0, stream>>>(src, dst, dinv, x, tx);
    wmma_gemm_kernel<true, false, true, false, IN_DIM><<<dim3(gx, (H1 + 15) / 16), gblk, 0, stream>>>(
        x, W1, tx, W1 + IN_DIM * H1, b1, nullptr, h, N_NODES, H1, H1);

    // ---- layer 2: x1 = relu(h@W2[0] + tx@W2[1] + b2) ----
    zero4_kernel<<<(N_NODES * H1 / 4 + BT - 1) / BT, BT, 0, stream>>>(tx, N_NODES * H1 / 4);
    scatter_kernel<H1><<<(NE * (H1 / 4) + BT - 1) / BT, BT, 0, stream>>>(src, dst, dinv, h, tx);
    wmma_gemm_kernel<true, false, true, false, H1><<<dim3(gx, (H2 + 15) / 16), gblk, 0, stream>>>(
        h, W2, tx, W2 + H1 * H2, b2, nullptr, x1, N_NODES, H2, H2);

    // ---- lin heads: x2 = x1 + relu(x@lin1_W^T + b), z = x1 + relu(x@lin2_W^T + b) ----
    wmma_gemm_kernel<false, true, true, true, IN_DIM><<<dim3(gx, (H2 + 15) / 16), gblk, 0, stream>>>(
        x, l1W, nullptr, nullptr, l1b, x1, x2, N_NODES, H2, H2);
    wmma_gemm_kernel<false, true, true, true, IN_DIM><<<dim3(gx, (H2 + 15) / 16), gblk, 0, stream>>>(
        x, l2W, nullptr, nullptr, l2b, x1, z, N_NODES, H2, H2);

    // ---- link prediction loss ----
    loss_kernel<<<(NP + BT - 1) / BT, BT, 0, stream>>>(epos, epos + NP, z, acc, NP, 0);
    loss_kernel<<<(NP + BT - 1) / BT, BT, 0, stream>>>(eneg, eneg + NP, z, acc, NP, 1);

    // ---- layer 3: out = x2@W3[0] + tx@W3[1] + b3 (no relu) ----
    zero4_kernel<<<(N_NODES * H2 / 4 + BT - 1) / BT, BT, 0, stream>>>(tx, N_NODES * H2 / 4);
    scatter_kernel<H2><<<(NE * (H2 / 4) + BT - 1) / BT, BT, 0, stream>>>(src, dst, dinv, x2, tx);
    wmma_gemm_kernel<true, false, false, false, H2><<<dim3(gx, 1), gblk, 0, stream>>>(
        x2, W3, tx, W3 + H2 * OUT_DIM, b3, nullptr, out, N_NODES, OUT_DIM, OUT_DIM);

    // ---- tail: r_loss, c1, c2 ----
    finalize_kernel<<<1, 1, 0, stream>>>(acc, c1, c2, out + N_NODES * OUT_DIM);
}